// OptimisedRNNFFT_78941498901169
// MI455X (gfx1250) — compile-verified
//
#include <hip/hip_runtime.h>

typedef __attribute__((ext_vector_type(2))) float v2f;
typedef __attribute__((ext_vector_type(4))) float v4f;
typedef __attribute__((ext_vector_type(8))) float v8f;

#define ROW_N 4096
#define BSTR 17              // 16 + 1 pad: conflict-free row AND column LDS access
#define ASTR (16 * BSTR)     // 272 floats per a-block
#define CUBE (16 * ASTR)     // 4352 floats per row cube (17,408 B)

// D(16x16,f32) = A(16x4,f32) x B(4x16,f32) + C ; chained 4x for K=16.
// Operand packing (ISA 7.12.2, f32):
//   A: lanes 0-15 hold M=lo; vgpr pair holds K = 4k+2*hi, 4k+2*hi+1
//   B: lanes hold N=lo;      vgpr pair holds K rows 4k+2*hi, 4k+2*hi+1
//   C/D: lane holds col N=lo, rows M = v + 8*hi (v = 0..7)
//
// One row (16x16x16 cube) per 128-thread block (4 waves):
//   stage2+1: wave w owns a-tiles {w, w+4, w+8, w+12}, fused, LDS-transposed
//   stage0:   wave w owns n-tiles {w, w+4, w+8, w+12}
// Residual of stage2 folded into the butterfly matrix: t2 = X_a * (M2 + I).
// NOTE: all per-lane operand tables are held in v2f[4] indexed ONLY by the
// compile-time-unrolled k (dynamic indexing of private arrays forces
// cndmask select ladders on gfx1250 — avoided deliberately).
__global__ __launch_bounds__(128)
void rnnfft_butterfly_kernel(const float* __restrict__ x,
                             const float* __restrict__ w0,
                             const float* __restrict__ w1,
                             const float* __restrict__ w2,
                             const float* __restrict__ l0,
                             const float* __restrict__ l1,
                             const float* __restrict__ l2,
                             float* __restrict__ out)
{
    __shared__ float xbuf[CUBE];   // original X (kept for stage-0 residual)
    __shared__ float tbuf[CUBE];   // T2 then T1 (in place per a-block)

    const int tid  = threadIdx.x;
    const int lane = tid & 31;
    const int wave = tid >> 5;
    const int lo   = lane & 15;
    const int hi   = lane >> 4;

    const float* __restrict__ xr   = x   + (size_t)blockIdx.x * ROW_N;
    float* __restrict__       outr = out + (size_t)blockIdx.x * ROW_N;

    // ---------- stage-in: 128 threads, coalesced b128 loads -> padded LDS cube
#pragma unroll
    for (int t = 0; t < 8; ++t) {
        int off = t * 512 + tid * 4;
        v4f v = *(const v4f*)(xr + off);
        int a = off >> 8, rem = off & 255;
        int b = rem >> 4, j = rem & 15;
        float* p = xbuf + a * ASTR + b * BSTR + j;
        p[0] = v.x; p[1] = v.y; p[2] = v.z; p[3] = v.w;
    }

    // ---------- preload small operands (L0/L2-resident, reused by every tile)
    v2f b2[4], a1[4], a0[4], w1v[4];
#pragma unroll
    for (int k = 0; k < 4; ++k) {          // k compile-time; j0 only feeds addresses
        int j0 = 4 * k + 2 * hi;
        // stage-2 B operand with residual folded: M2'[j][i] = w2[j]*l2[i][j] + (i==j)
        b2[k].x = w2[j0]     * l2[lo * 16 + j0]     + ((j0     == lo) ? 1.0f : 0.0f);
        b2[k].y = w2[j0 + 1] * l2[lo * 16 + j0 + 1] + ((j0 + 1 == lo) ? 1.0f : 0.0f);
        a1[k].x = l1[lo * 16 + j0];  a1[k].y = l1[lo * 16 + j0 + 1];
        a0[k].x = l0[lo * 16 + j0];  a0[k].y = l0[lo * 16 + j0 + 1];
        // stage-1 weights, loaded straight to the chunk registers (no w1c[16] array)
        w1v[k].x = w1[j0 * 16 + lo];
        w1v[k].y = w1[(j0 + 1) * 16 + lo];
    }

    __syncthreads();   // s_barrier_signal/s_barrier_wait

    // ---------- stages 2+1 fused: T1_a = L1 * (w1 ⊙ (X_a * (M2 + I)))
#pragma unroll
    for (int t = 0; t < 4; ++t) {
        int a = wave + 4 * t;                      // wave-uniform tile id
        const float* xb = xbuf + a * ASTR;
        float* tb = tbuf + a * ASTR;

        v8f acc = {};                              // residual folded -> C = 0
#pragma unroll
        for (int k = 0; k < 4; ++k) {
            int j0 = 4 * k + 2 * hi;
            v2f av;
            av.x = xb[lo * BSTR + j0];
            av.y = xb[lo * BSTR + j0 + 1];
            acc = __builtin_amdgcn_wmma_f32_16x16x4_f32(
                      false, av, false, b2[k], (short)0, acc, false, false);
        }
#pragma unroll
        for (int v = 0; v < 8; ++v)                // T2_a -> LDS (acts as transpose)
            tb[(v + 8 * hi) * BSTR + lo] = acc[v];

        v8f acc2 = {};                             // stage 1: no residual
#pragma unroll
        for (int k = 0; k < 4; ++k) {
            int j0 = 4 * k + 2 * hi;
            v2f bv;                                // same-wave DS is in-order: sees T2 stores
            bv.x = w1v[k].x * tb[j0 * BSTR + lo];
            bv.y = w1v[k].y * tb[(j0 + 1) * BSTR + lo];
            acc2 = __builtin_amdgcn_wmma_f32_16x16x4_f32(
                       false, a1[k], false, bv, (short)0, acc2, false, false);
        }
#pragma unroll
        for (int v = 0; v < 8; ++v)                // T1_a overwrites in place
            tb[(v + 8 * hi) * BSTR + lo] = acc2[v];
    }

    __syncthreads();   // stage 0 reads T1 across all waves' a-blocks

    // ---------- stage 0: out[:, n-blk] = L0 * (w0 ⊙ T1)[:, n-blk] + X[:, n-blk]
#pragma unroll
    for (int t = 0; t < 4; ++t) {
        int n = wave + 4 * t;                      // wave-uniform tile id
        v8f acc;
#pragma unroll
        for (int v = 0; v < 8; ++v)                // C = residual X from LDS copy
            acc[v] = xbuf[(v + 8 * hi) * ASTR + n * BSTR + lo];
#pragma unroll
        for (int k = 0; k < 4; ++k) {
            int j0 = 4 * k + 2 * hi;
            v2f bv;
            bv.x = w0[j0 * 256 + n * 16 + lo] *
                   tbuf[j0 * ASTR + n * BSTR + lo];
            bv.y = w0[(j0 + 1) * 256 + n * 16 + lo] *
                   tbuf[(j0 + 1) * ASTR + n * BSTR + lo];
            acc = __builtin_amdgcn_wmma_f32_16x16x4_f32(
                      false, a0[k], false, bv, (short)0, acc, false, false);
        }
#pragma unroll
        for (int v = 0; v < 8; ++v)                // coalesced 64B-per-half-wave stores
            outr[(size_t)(v + 8 * hi) * 256 + n * 16 + lo] = acc[v];
    }
}

extern "C" void kernel_launch(void* const* d_in, const int* in_sizes, int n_in,
                              void* d_out, int out_size, void* d_ws, size_t ws_size,
                              hipStream_t stream) {
    (void)n_in; (void)out_size; (void)d_ws; (void)ws_size;
    const float* x  = (const float*)d_in[0];
    const float* w0 = (const float*)d_in[1];
    const float* w1 = (const float*)d_in[2];
    const float* w2 = (const float*)d_in[3];
    const float* l0 = (const float*)d_in[4];
    const float* l1 = (const float*)d_in[5];
    const float* l2 = (const float*)d_in[6];
    float* out = (float*)d_out;

    int nrows = in_sizes[0] / ROW_N;               // 8192 -> one row per block
    hipLaunchKernelGGL(rnnfft_butterfly_kernel, dim3(nrows), dim3(128), 0, stream,
                       x, w0, w1, w2, l0, l1, l2, out);
}